// FastHadamardTransform_57226144252247
// MI455X (gfx1250) — compile-verified
//
#include <hip/hip_runtime.h>

// FWHT-4096 via H16 (x) H16 (x) H16 Kronecker decomposition on CDNA5 WMMA.
// One wave32 per 4096-float row; row staged in a private 16KB LDS segment.
// 3 passes x 16 tiles x 4 chained V_WMMA_F32_16X16X4_F32 per tile.

typedef __attribute__((ext_vector_type(2))) float v2f;
typedef __attribute__((ext_vector_type(4))) float v4f;
typedef __attribute__((ext_vector_type(8))) float v8f;

#define FWHT_DIM 4096
#define WAVES_PER_BLOCK 4
#define BLOCK_THREADS (WAVES_PER_BLOCK * 32)

__device__ __forceinline__ v8f wmma_16x16x4_f32(v2f a, v2f b, v8f c) {
    // 8 args: (neg_a, A, neg_b, B, c_mod, C, reuse_a, reuse_b)
    return __builtin_amdgcn_wmma_f32_16x16x4_f32(
        false, a, false, b, (short)0, c, false, false);
}

// One Hadamard pass over the 16x16x16 view of the row held in `seg`.
// Tile element (t, n, k) lives at seg[t*ST + n*SN + k*SK].
// Computes D[m][n] = sum_k H16[m][k] * X[k][n] for each of the 16 tiles,
// in place (full tile is read into registers before the stores).
template <int SK, int SN, int ST>
__device__ __forceinline__ void hadamard_pass(float* seg, const v2f* a, int lane) {
    const int n  = lane & 15;
    const int hi = lane >> 4;   // half-wave select
    const int ko = hi * 2;      // K offset inside each 4-chunk (B layout)
    const int mo = hi * 8;      // D row offset (C/D layout: vgpr i -> M=i / i+8)

    #pragma unroll
    for (int t = 0; t < 16; ++t) {
        float* base = seg + t * ST + n * SN;

        v2f b[4];
        #pragma unroll
        for (int j = 0; j < 4; ++j) {
            const int k0 = 4 * j + ko;
            b[j].x = base[(k0 + 0) * SK];
            b[j].y = base[(k0 + 1) * SK];
        }

        v8f acc = {};
        #pragma unroll
        for (int j = 0; j < 4; ++j)
            acc = wmma_16x16x4_f32(a[j], b[j], acc);

        #pragma unroll
        for (int i = 0; i < 8; ++i)
            base[(i + mo) * SK] = acc[i];
    }
}

__global__ void __launch_bounds__(BLOCK_THREADS)
FastHadamardTransform_57226144252247_kernel(const float* __restrict__ in,
                                            float* __restrict__ out,
                                            int nrows) {
    __shared__ __align__(16) float lds[WAVES_PER_BLOCK * FWHT_DIM];

    const int lane = threadIdx.x & 31;
    const int wave = threadIdx.x >> 5;
    const int row  = blockIdx.x * WAVES_PER_BLOCK + wave;
    if (row >= nrows) return;  // wave-uniform; surviving waves keep EXEC all-ones

    float* seg = lds + wave * FWHT_DIM;
    const float* gin = in + (size_t)row * FWHT_DIM;

    // Global -> LDS: 32 iterations of per-lane float4 (512B/wave contiguous).
    #pragma unroll 8
    for (int i = 0; i < FWHT_DIM / 128; ++i) {
        v4f v = ((const v4f*)gin)[i * 32 + lane];
        ((v4f*)seg)[i * 32 + lane] = v;
    }

    // Build A = H16 once: A[m][k] = (-1)^popcount(m&k).
    // 16x4 f32 A layout: lanes 0-15 -> M=lane, VGPR0/1 = K0/K1;
    // lanes 16-31 -> same M, K2/K3.
    const int m  = lane & 15;
    const int ko = (lane >> 4) * 2;
    v2f a[4];
    #pragma unroll
    for (int j = 0; j < 4; ++j) {
        const int k0 = 4 * j + ko;
        a[j].x = (__builtin_popcount(m & (k0 + 0)) & 1) ? -1.0f : 1.0f;
        a[j].y = (__builtin_popcount(m & (k0 + 1)) & 1) ? -1.0f : 1.0f;
    }

    // Row index i = c2*256 + c1*16 + c0; transform each 16-ary digit.
    hadamard_pass<1,   16, 256>(seg, a, lane);  // c0: tile=c2, col=c1
    hadamard_pass<16,  1,  256>(seg, a, lane);  // c1: tile=c2, col=c0
    hadamard_pass<256, 1,  16 >(seg, a, lane);  // c2: tile=c1, col=c0

    // LDS -> global with normalization 2^-(L(L+1)/4), L=12 -> 2^-39.
    const float scale = 0x1.0p-39f;
    float* gout = out + (size_t)row * FWHT_DIM;
    #pragma unroll 8
    for (int i = 0; i < FWHT_DIM / 128; ++i) {
        v4f v = ((v4f*)seg)[i * 32 + lane];
        ((v4f*)gout)[i * 32 + lane] = v * scale;
    }
}

extern "C" void kernel_launch(void* const* d_in, const int* in_sizes, int n_in,
                              void* d_out, int out_size, void* d_ws, size_t ws_size,
                              hipStream_t stream) {
    (void)n_in; (void)d_ws; (void)ws_size; (void)out_size;
    const float* x = (const float*)d_in[0];
    float* y = (float*)d_out;
    const int nrows = in_sizes[0] / FWHT_DIM;          // 8192 for 4x2048x4096
    const int grid  = (nrows + WAVES_PER_BLOCK - 1) / WAVES_PER_BLOCK;
    FastHadamardTransform_57226144252247_kernel<<<grid, BLOCK_THREADS, 0, stream>>>(
        x, y, nrows);
}